// MaskRCNN_73478300500553
// MI455X (gfx1250) — compile-verified
//
#include <hip/hip_runtime.h>

#define ROI_OUT 14
#define NPOS    (ROI_OUT * ROI_OUT)   // 196
#define NCH     256
#define WORK    (NCH * NPOS)          // 50176 = 196 * 256 exactly

static __device__ __forceinline__ int imin(int a, int b) { return a < b ? a : b; }
static __device__ __forceinline__ int imax(int a, int b) { return a > b ? a : b; }

static __device__ __forceinline__ void split_coord(float coord, int size,
                                                   int& c0, int& c1, float& w) {
    float c = fminf(fmaxf(coord, 0.0f), (float)(size - 1));
    float f = floorf(c);
    c0 = (int)f;
    c1 = imin(c0 + 1, size - 1);
    w  = c - f;
}

__global__ __launch_bounds__(256) void roialign_fpn_kernel(
    const float* __restrict__ p2, const float* __restrict__ p3,
    const float* __restrict__ p4, const float* __restrict__ p5,
    const float* __restrict__ rois, float* __restrict__ out)
{
    const int r = blockIdx.x;
    const int t = threadIdx.x;

    // Per-position sample descriptors, built once per RoI.
    __shared__ int4   soff[NPOS];   // element offsets of 4 bilinear corners in a channel plane
    __shared__ float2 sw[NPOS];     // (wx, wy)

    // Every thread redundantly derives the RoI's level; wave-uniform, so the
    // compiler scalarizes this whole block (s_load + SALU, confirmed in asm).
    const float* roi = rois + (size_t)r * 5;
    const int   bidx = (int)roi[0];
    const float bx1 = roi[1], by1 = roi[2], bx2 = roi[3], by2 = roi[4];
    const float bw = fabsf(bx2 - bx1), bh = fabsf(by2 - by1);
    int lvl = (int)floorf(4.0f + log2f(sqrtf(bw * bh) * (1.0f / 224.0f))) - 2;
    lvl = imax(0, imin(3, lvl));

    const int   H  = (lvl == 0) ? 256 : (lvl == 1) ? 128 : (lvl == 2) ? 64 : 32;
    const int   W  = H;
    const float sc = (lvl == 0) ? 0.25f : (lvl == 1) ? 0.125f
                   : (lvl == 2) ? 0.0625f : 0.03125f;
    const float* base = (lvl == 0) ? p2 : (lvl == 1) ? p3 : (lvl == 2) ? p4 : p5;

    if (t < NPOS) {
        const int px = t % ROI_OUT;
        const int py = t / ROI_OUT;
        const float fx1 = bx1 * sc, fy1 = by1 * sc;
        const float binw = (bx2 * sc - fx1) * (1.0f / (float)ROI_OUT);
        const float binh = (by2 * sc - fy1) * (1.0f / (float)ROI_OUT);
        const float gx = fx1 + ((float)px + 0.5f) * binw;
        const float gy = fy1 + ((float)py + 0.5f) * binh;
        int ix0, ix1, iy0, iy1; float wx, wy;
        split_coord(gx, W, ix0, ix1, wx);
        split_coord(gy, H, iy0, iy1, wy);
        soff[t] = make_int4(iy0 * W + ix0, iy0 * W + ix1,
                            iy1 * W + ix0, iy1 * W + ix1);
        sw[t]   = make_float2(wx, wy);
    }
    __syncthreads();

    const int    HW = H * W;
    const float* fb = base + (size_t)bidx * NCH * HW;   // batch's pyramid level
    float*       op = out + (size_t)r * WORK;

    // Flat loop: i = c*196 + p. Consecutive threads -> consecutive output
    // elements (coalesced NT stores) and adjacent sample bins (clustered
    // gathers over ~1-2 channel planes per wave-iteration). Unroll 4 gives
    // each thread 16 independent outstanding gathers for latency hiding.
    #pragma unroll 4
    for (int i = t; i < WORK; i += 256) {
        const unsigned c = (unsigned)i / (unsigned)NPOS;   // const-divisor -> mul/shift
        const int      p = i - (int)c * NPOS;

        const int4   o = soff[p];
        const float2 w = sw[p];
        const float* fc = fb + (size_t)c * HW;

        // Branchless prefetch of the next-but-one channel plane of this patch
        // (gfx1250 global_prefetch_b8): the pyramid lives in the 192MB L2;
        // this hides the L2 -> WGP$ hop. Clamped address instead of a branch.
        const unsigned cpf = (c + 2u < (unsigned)NCH) ? (c + 2u) : (unsigned)(NCH - 1);
        __builtin_prefetch(fb + (size_t)cpf * HW + o.x, 0, 3);

        const float v00 = fc[o.x];
        const float v01 = fc[o.y];
        const float v10 = fc[o.z];
        const float v11 = fc[o.w];

        const float top = v00 + w.x * (v01 - v00);
        const float bot = v10 + w.x * (v11 - v10);
        const float v   = top + w.y * (bot - top);

        // Streamed output (205MB, never re-read): non-temporal store so the
        // feature pyramid (179MB) stays resident in the 192MB L2 across the
        // harness's graph replays.
        __builtin_nontemporal_store(v, op + i);
    }
}

extern "C" void kernel_launch(void* const* d_in, const int* in_sizes, int n_in,
                              void* d_out, int out_size, void* d_ws, size_t ws_size,
                              hipStream_t stream) {
    (void)n_in; (void)d_ws; (void)ws_size; (void)out_size;
    const float* p2   = (const float*)d_in[0];
    const float* p3   = (const float*)d_in[1];
    const float* p4   = (const float*)d_in[2];
    const float* p5   = (const float*)d_in[3];
    const float* rois = (const float*)d_in[4];
    const int R = in_sizes[4] / 5;   // 1024 RoIs

    roialign_fpn_kernel<<<dim3(R), dim3(256), 0, stream>>>(
        p2, p3, p4, p5, rois, (float*)d_out);
}